// MambaClass_63720134803785
// MI455X (gfx1250) — compile-verified
//
#include <hip/hip_runtime.h>
#include <hip/hip_bf16.h>
#include <math.h>

#define NB 8
#define LSEQ 250
#define DMODEL 512
#define DINNER 1024
#define DSTATE 64
#define DTRANK 32
#define BL 2000          // NB * LSEQ
#define FLATK 120000

typedef float v2f __attribute__((ext_vector_type(2)));
typedef float v8f __attribute__((ext_vector_type(8)));

__device__ __forceinline__ v8f wmma4(v2f a, v2f b, v8f c) {
    return __builtin_amdgcn_wmma_f32_16x16x4_f32(
        false, a, false, b, (short)0, c, false, false);
}

// ---------------------------------------------------------------------------
// Generic fp32 WMMA GEMM:  C[M,N] = A[M,K] * W[N,K]^T  (+bias)(+act)
// One wave computes a 32x32 block (2x2 tiles of 16x16) via V_WMMA_F32_16X16X4_F32.
// Out-of-range rows are clamped for loads (garbage only reaches guarded outputs).
// mode: 0 = store (+bias), 1 = store (+bias, softplus), 2 = split-K atomicAdd
// ---------------------------------------------------------------------------
__global__ void gemm16(const float* __restrict__ A, int lda,
                       const float* __restrict__ W, int ldb,
                       float* __restrict__ C, int ldc,
                       const float* __restrict__ bias,
                       int M, int N, int K, int kChunk, int mode)
{
    int lane = threadIdx.x & 31;
    int half = lane >> 4;        // 0: K pair {0,1}, 1: K pair {2,3}
    int l16  = lane & 15;
    int nt = blockIdx.x, mt = blockIdx.y;
    int m0 = mt * 32 + l16,  m1 = m0 + 16;   // A rows this lane loads
    int n0 = nt * 32 + l16,  n1 = n0 + 16;   // W rows (= B cols) this lane loads
    int m0c = (m0 < M) ? m0 : M - 1;
    int m1c = (m1 < M) ? m1 : M - 1;
    int n0c = (n0 < N) ? n0 : N - 1;
    int n1c = (n1 < N) ? n1 : N - 1;

    long k0 = (long)blockIdx.z * kChunk;
    long kend = k0 + kChunk; if (kend > K) kend = K;
    long nk = kend - k0;

    const float* a0p = A + (size_t)m0c * lda + 2 * half + k0;
    const float* a1p = A + (size_t)m1c * lda + 2 * half + k0;
    const float* b0p = W + (size_t)n0c * ldb + 2 * half + k0;
    const float* b1p = W + (size_t)n1c * ldb + 2 * half + k0;

    v8f acc00 = {0.f,0.f,0.f,0.f,0.f,0.f,0.f,0.f};
    v8f acc01 = acc00, acc10 = acc00, acc11 = acc00;

    long k = 0;
    for (; k + 8 <= nk; k += 8) {
        v2f a0 = *(const v2f*)(a0p + k);
        v2f a1 = *(const v2f*)(a1p + k);
        v2f b0 = *(const v2f*)(b0p + k);
        v2f b1 = *(const v2f*)(b1p + k);
        v2f a0u = *(const v2f*)(a0p + k + 4);
        v2f a1u = *(const v2f*)(a1p + k + 4);
        v2f b0u = *(const v2f*)(b0p + k + 4);
        v2f b1u = *(const v2f*)(b1p + k + 4);
        acc00 = wmma4(a0, b0, acc00);
        acc01 = wmma4(a0, b1, acc01);
        acc10 = wmma4(a1, b0, acc10);
        acc11 = wmma4(a1, b1, acc11);
        acc00 = wmma4(a0u, b0u, acc00);
        acc01 = wmma4(a0u, b1u, acc01);
        acc10 = wmma4(a1u, b0u, acc10);
        acc11 = wmma4(a1u, b1u, acc11);
    }
    for (; k < nk; k += 4) {        // tail (unused by current launches)
        v2f a0 = *(const v2f*)(a0p + k);
        v2f a1 = *(const v2f*)(a1p + k);
        v2f b0 = *(const v2f*)(b0p + k);
        v2f b1 = *(const v2f*)(b1p + k);
        acc00 = wmma4(a0, b0, acc00);
        acc01 = wmma4(a0, b1, acc01);
        acc10 = wmma4(a1, b0, acc10);
        acc11 = wmma4(a1, b1, acc11);
    }

    // epilogue: 4 tiles, store-side guards
    #pragma unroll
    for (int tn = 0; tn < 2; ++tn) {
        int n = tn ? n1 : n0;
        if (n >= N) continue;
        float bb = (mode != 2 && bias) ? bias[n] : 0.f;
        #pragma unroll
        for (int tm = 0; tm < 2; ++tm) {
            v8f acc = tm ? (tn ? acc11 : acc10) : (tn ? acc01 : acc00);
            int mbase = mt * 32 + tm * 16 + half * 8;   // C/D: vgpr r -> M=r (+8 hi half)
            #pragma unroll
            for (int r = 0; r < 8; ++r) {
                int mm = mbase + r;
                if (mm < M) {
                    if (mode == 2) {
                        atomicAdd(&C[(size_t)mm * ldc + n], acc[r]);
                    } else {
                        float v = acc[r] + bb;
                        if (mode == 1) v = (v > 20.f) ? v : log1pf(expf(v));  // softplus
                        C[(size_t)mm * ldc + n] = v;
                    }
                }
            }
        }
    }
}

// ---------------------------------------------------------------------------
// u[bl, c] = x[b, t*480 + c] (c < 480) else 0     (pad 480 -> 512)
// ---------------------------------------------------------------------------
__global__ void prep_u(const float* __restrict__ x, float* __restrict__ u)
{
    int idx = blockIdx.x * blockDim.x + threadIdx.x;
    if (idx >= BL * DMODEL) return;
    int c = idx & (DMODEL - 1);
    int bl = idx >> 9;
    int b = bl / LSEQ, t = bl % LSEQ;
    u[idx] = (c < 480) ? x[(size_t)b * FLATK + (size_t)t * 480 + c] : 0.f;
}

// As2[d,s] = -exp(A_log[d,s]) * log2(e)   (so scan can use exp2f)
__global__ void prep_as(const float* __restrict__ A_log, float* __restrict__ As2)
{
    int idx = blockIdx.x * blockDim.x + threadIdx.x;
    if (idx >= DINNER * DSTATE) return;
    As2[idx] = -expf(A_log[idx]) * 1.4426950408889634f;
}

// depthwise causal conv (width 4) + bias + silu; input = first half of xz
__global__ void conv_silu(const float* __restrict__ xz,
                          const float* __restrict__ conv_w,
                          const float* __restrict__ conv_b,
                          float* __restrict__ xc)
{
    int idx = blockIdx.x * blockDim.x + threadIdx.x;
    if (idx >= BL * DINNER) return;
    int d = idx & (DINNER - 1);
    int bl = idx >> 10;
    int l = bl % LSEQ;
    float acc = conv_b[d];
    #pragma unroll
    for (int k = 0; k < 4; ++k) {
        int ls = l - 3 + k;
        if (ls >= 0)
            acc = fmaf(xz[(size_t)(bl - 3 + k) * 2048 + d], conv_w[d * 4 + k], acc);
    }
    xc[idx] = acc / (1.f + expf(-acc));     // silu
}

// ---------------------------------------------------------------------------
// Selective scan. Thread = (b, d, state-octet). 8 states/thread in registers.
// Block = 256 threads = 32 d's of one batch; B/C/dt/u staged via LDS per step.
// Epilogue fused: ys = (y + u*D) * silu(z)
// ---------------------------------------------------------------------------
__global__ void scan_kernel(const float* __restrict__ xz,    // [BL, 2048] (z in cols 1024..)
                            const float* __restrict__ xc,    // [BL, 1024]
                            const float* __restrict__ xdbl,  // [BL, 160]  (B: 32..95, C: 96..159)
                            const float* __restrict__ dtb,   // [BL, 1024] softplus'd dt
                            const float* __restrict__ As2,   // [1024, 64]
                            const float* __restrict__ Dvec,  // [1024]
                            float* __restrict__ ys)          // [BL, 1024]
{
    __shared__ float sB[DSTATE], sC[DSTATE], sdt[32], su[32];
    int t = threadIdx.x;           // 0..255
    int lane = t & 31;
    int w = t >> 5;                // wave 0..7
    int dloc = lane & 3;
    int sg = lane >> 2;            // state octet 0..7
    int s0 = sg * 8;
    int b = blockIdx.y;
    int dbase = blockIdx.x * 32;
    int dl = w * 4 + dloc;         // 0..31
    int d = dbase + dl;

    float as[8], h[8];
    #pragma unroll
    for (int i = 0; i < 8; ++i) {
        as[i] = As2[(size_t)d * DSTATE + s0 + i];
        h[i] = 0.f;
    }
    float Dd = Dvec[d];

    for (int l = 0; l < LSEQ; ++l) {
        int bl = b * LSEQ + l;
        if (t < 64)       sB[t]        = xdbl[(size_t)bl * 160 + 32 + t];
        else if (t < 128) sC[t - 64]   = xdbl[(size_t)bl * 160 + 96 + (t - 64)];
        else if (t < 160) sdt[t - 128] = dtb[(size_t)bl * DINNER + dbase + (t - 128)];
        else if (t < 192) su[t - 160]  = xc[(size_t)bl * DINNER + dbase + (t - 160)];
        __syncthreads();

        float dt = sdt[dl], u = su[dl];
        float dtu = dt * u;
        float yp = 0.f;
        #pragma unroll
        for (int i = 0; i < 8; ++i) {
            float dA = exp2f(dt * as[i]);              // exp(dt * A)
            h[i] = fmaf(h[i], dA, dtu * sB[s0 + i]);
            yp = fmaf(h[i], sC[s0 + i], yp);
        }
        // reduce y over the 8 state octets (lanes xor 4, 8, 16)
        yp += __shfl_xor(yp, 4);
        yp += __shfl_xor(yp, 8);
        yp += __shfl_xor(yp, 16);
        if (sg == 0) {
            float z = xz[(size_t)bl * 2048 + DINNER + d];
            float sz = z / (1.f + expf(-z));           // silu(z)
            ys[(size_t)bl * DINNER + d] = (yp + u * Dd) * sz;
        }
        __syncthreads();
    }
}

// selu + transpose scatter: feat[b, f*5000 + l*20 + p] = selu(yo[b*250+l, p*24+f])
__global__ void selu_scatter(const float* __restrict__ yo, float* __restrict__ feat)
{
    int idx = blockIdx.x * blockDim.x + threadIdx.x;
    if (idx >= NB * LSEQ * 480) return;
    int c = idx % 480;
    int r = idx / 480;
    int l = r % LSEQ;
    int b = r / LSEQ;
    float v = yo[(size_t)(b * LSEQ + l) * DMODEL + c];
    const float alpha = 1.6732632423543772f, scale = 1.0507009873554805f;
    v = scale * (v > 0.f ? v : alpha * (expf(v) - 1.f));
    int p = c / 24, f = c % 24;
    feat[(size_t)b * FLATK + f * 5000 + l * 20 + p] = v;
}

// out1[b, n] = fc1_b[n]  (split-K atomics accumulate on top)
__global__ void fc_init(const float* __restrict__ bias, float* __restrict__ o)
{
    int idx = blockIdx.x * blockDim.x + threadIdx.x;
    if (idx >= NB * 512) return;
    o[idx] = bias[idx & 511];
}

// ---------------------------------------------------------------------------
extern "C" void kernel_launch(void* const* d_in, const int* in_sizes, int n_in,
                              void* d_out, int out_size, void* d_ws, size_t ws_size,
                              hipStream_t stream)
{
    const float* x        = (const float*)d_in[0];
    const float* in_w     = (const float*)d_in[1];
    const float* conv_w   = (const float*)d_in[2];
    const float* conv_b   = (const float*)d_in[3];
    const float* xproj_w  = (const float*)d_in[4];
    const float* dtproj_w = (const float*)d_in[5];
    const float* dtproj_b = (const float*)d_in[6];
    const float* A_log    = (const float*)d_in[7];
    const float* Dvec     = (const float*)d_in[8];
    const float* out_w    = (const float*)d_in[9];
    const float* fc1_w    = (const float*)d_in[10];
    const float* fc1_b    = (const float*)d_in[11];
    const float* fc2_w    = (const float*)d_in[12];
    const float* fc2_b    = (const float*)d_in[13];
    const float* fc3_w    = (const float*)d_in[14];
    const float* fc3_b    = (const float*)d_in[15];
    const float* fc4_w    = (const float*)d_in[16];
    const float* fc4_b    = (const float*)d_in[17];

    char* ws = (char*)d_ws;
    size_t off = 0;
    auto alloc = [&](size_t elems) {
        size_t p = off;
        off += (elems * sizeof(float) + 255) & ~(size_t)255;
        return (float*)(ws + p);
    };
    float* u    = alloc((size_t)BL * DMODEL);
    float* xz   = alloc((size_t)BL * 2048);
    float* xc   = alloc((size_t)BL * DINNER);
    float* xdbl = alloc((size_t)BL * 160);
    float* dtb  = alloc((size_t)BL * DINNER);
    float* As2  = alloc((size_t)DINNER * DSTATE);
    float* ys   = alloc((size_t)BL * DINNER);
    float* yo   = alloc((size_t)BL * DMODEL);
    float* feat = alloc((size_t)NB * FLATK);
    float* o1   = alloc((size_t)NB * 512);
    float* o2   = alloc((size_t)NB * 256);
    float* o3   = alloc((size_t)NB * 64);
    (void)in_sizes; (void)n_in; (void)out_size; (void)ws_size;

    const int MT = (BL + 31) / 32;   // 63 M-blocks for the 2000-row GEMMs

    // 1. build padded input u [2000, 512]
    prep_u<<<(BL * DMODEL + 255) / 256, 256, 0, stream>>>(x, u);
    // 2. As2 table for the scan
    prep_as<<<(DINNER * DSTATE + 255) / 256, 256, 0, stream>>>(A_log, As2);
    // 3. in_proj: xz[2000,2048] = u @ in_w^T
    gemm16<<<dim3(2048 / 32, MT, 1), 32, 0, stream>>>(
        u, DMODEL, in_w, DMODEL, xz, 2048, nullptr, BL, 2048, DMODEL, DMODEL, 0);
    // 4. depthwise conv + silu -> xc[2000,1024]
    conv_silu<<<(BL * DINNER + 255) / 256, 256, 0, stream>>>(xz, conv_w, conv_b, xc);
    // 5. x_proj: xdbl[2000,160] = xc @ xproj_w^T
    gemm16<<<dim3(160 / 32, MT, 1), 32, 0, stream>>>(
        xc, DINNER, xproj_w, DINNER, xdbl, 160, nullptr, BL, 160, DINNER, DINNER, 0);
    // 6. dt_proj + softplus: dtb[2000,1024] (A = first 32 cols of xdbl, lda=160)
    gemm16<<<dim3(DINNER / 32, MT, 1), 32, 0, stream>>>(
        xdbl, 160, dtproj_w, DTRANK, dtb, DINNER, dtproj_b, BL, DINNER, DTRANK, DTRANK, 1);
    // 7. selective scan (fused +u*D and *silu(z)) -> ys[2000,1024]
    scan_kernel<<<dim3(DINNER / 32, NB), 256, 0, stream>>>(xz, xc, xdbl, dtb, As2, Dvec, ys);
    // 8. out_proj: yo[2000,512] = ys @ out_w^T
    gemm16<<<dim3(DMODEL / 32, MT, 1), 32, 0, stream>>>(
        ys, DINNER, out_w, DINNER, yo, DMODEL, nullptr, BL, DMODEL, DINNER, DINNER, 0);
    // 9. selu + transpose scatter -> feat[8, 120000]
    selu_scatter<<<(NB * LSEQ * 480 + 255) / 256, 256, 0, stream>>>(yo, feat);
    // 10. fc1: bias init then split-K WMMA with atomic accumulation (245.8 MB stream)
    fc_init<<<(NB * 512 + 255) / 256, 256, 0, stream>>>(fc1_b, o1);
    gemm16<<<dim3(512 / 32, 1, 250), 32, 0, stream>>>(
        feat, FLATK, fc1_w, FLATK, o1, 512, nullptr, NB, 512, FLATK, 480, 2);
    // 11-13. fc2/fc3/fc4
    gemm16<<<dim3(256 / 32, 1, 1), 32, 0, stream>>>(
        o1, 512, fc2_w, 512, o2, 256, fc2_b, NB, 256, 512, 512, 0);
    gemm16<<<dim3(64 / 32, 1, 1), 32, 0, stream>>>(
        o2, 256, fc3_w, 256, o3, 64, fc3_b, NB, 64, 256, 256, 0);
    gemm16<<<dim3(1, 1, 1), 32, 0, stream>>>(
        o3, 64, fc4_w, 64, (float*)d_out, 8, fc4_b, NB, 8, 64, 64, 0);
}